// TensorProduct_5231270166734
// MI455X (gfx1250) — compile-verified
//
#include <hip/hip_runtime.h>

typedef __attribute__((ext_vector_type(2))) float v2f;
typedef __attribute__((ext_vector_type(8))) float v8f;

#define NUM_ORDERS 16
#define C_DIM 64
#define K_DENSE 256  // 16*16 (m1,m2) pairs, densified CG matrix width

// ---------------------------------------------------------------------------
// Kernel 1: densify the COO CG table into a 16 x 256 row-major matrix in d_ws.
// A[m][m1*16+m2] = CG(m, m1, m2).  (Mseg,M1,M2) triples are unique (the flat
// index l*l+l+m encodes (l,m) uniquely), so plain stores suffice.
// ---------------------------------------------------------------------------
__global__ void build_cg_dense(const float* __restrict__ vals,
                               const int* __restrict__ M1,
                               const int* __restrict__ M2,
                               const int* __restrict__ Mseg,
                               float* __restrict__ A, int nnz) {
  const int t = threadIdx.x;  // single block of 512 threads
  for (int i = t; i < NUM_ORDERS * K_DENSE; i += blockDim.x) A[i] = 0.0f;
  __syncthreads();
  for (int i = t; i < nnz; i += blockDim.x) {
    A[Mseg[i] * K_DENSE + M1[i] * NUM_ORDERS + M2[i]] = vals[i];
  }
}

// ---------------------------------------------------------------------------
// Kernel 2: out[n,:,c0:c0+16] = A(16x256) @ Z(256x16),
//           Z[m1*16+m2, c] = x1[n,m1,c] * x2[n,m2,c]  (register outer product)
//
// One wave32 handles TWO batch rows (same channel tile): the LDS-resident A
// fragment (ds_load_b64) is shared by two independent WMMA accumulator
// chains, halving fragment traffic per tile and keeping the XDL pipe fed
// with independent v_wmma_f32_16x16x4_f32 ops.
//
// Fragment layouts (CDNA5 ISA 7.12.2), half = lane>>4:
//   A 16x4 : lane holds row M = lane&15;   vgpr j -> K = j + 2*half
//   B 4x16 : lane holds col N = lane&15;   vgpr j -> K = j + 2*half
//   C/D    : lane holds col N = lane&15;   vgpr v -> M = v + 8*half
// ---------------------------------------------------------------------------
__global__ void __launch_bounds__(256)
tp_wmma_kernel(const float* __restrict__ x1, const float* __restrict__ x2,
               const float* __restrict__ A, float* __restrict__ out,
               int n_batch) {
  __shared__ float Alds[NUM_ORDERS * K_DENSE];  // 16 KB CG matrix, block-shared

  // Cooperative stage: 256 threads x 4 x float4 = 16 KB (b128 load + ds b128).
  {
    const float4* src = (const float4*)A;
    float4* dst = (float4*)Alds;
#pragma unroll
    for (int i = 0; i < 4; ++i)
      dst[threadIdx.x + i * 256] = src[threadIdx.x + i * 256];
  }
  __syncthreads();

  const int lane = threadIdx.x & 31;
  const int wave = (blockIdx.x * blockDim.x + threadIdx.x) >> 5;
  const int pair = wave >> 2;  // pair of batch rows
  const int ct   = wave & 3;   // 16-wide channel tile
  const int n0   = pair * 2;
  if (n0 >= n_batch) return;   // wave-uniform: EXEC stays all-ones
  // Odd tail: duplicate the row (same values stored twice -> deterministic).
  const int n1 = (n0 + 1 < n_batch) ? n0 + 1 : n0;

  const int half = lane >> 4;
  const int col  = lane & 15;
  const int c    = ct * 16 + col;

  const float* p1a = x1 + (size_t)n0 * NUM_ORDERS * C_DIM + c;
  const float* p2a = x2 + (size_t)n0 * NUM_ORDERS * C_DIM + c;
  const float* p1b = x1 + (size_t)n1 * NUM_ORDERS * C_DIM + c;
  const float* p2b = x2 + (size_t)n1 * NUM_ORDERS * C_DIM + c;

  // Per-lane stationary operands: x1 over all 16 orders; x2 only at the 8
  // order indices this half-wave ever needs: 4*(t>>1) + (t&1) + 2*half.
  float w1a[16], w1b[16];
#pragma unroll
  for (int j = 0; j < 16; ++j) {
    w1a[j] = p1a[j * C_DIM];
    w1b[j] = p1b[j * C_DIM];
  }
  float w2a[8], w2b[8];
#pragma unroll
  for (int t = 0; t < 8; ++t) {
    const int o = (4 * (t >> 1) + (t & 1) + 2 * half) * C_DIM;
    w2a[t] = p2a[o];
    w2b[t] = p2b[o];
  }

  // This lane's slice of the CG matrix in LDS: row (lane&15), K offset 2*half.
  const float* arow = Alds + col * K_DENSE + 2 * half;

  v8f acc0 = {};
  v8f acc1 = {};
#pragma unroll
  for (int kk = 0; kk < 64; ++kk) {
    // A fragment: K = 4*kk + 2*half + {0,1}; adjacent + 8B aligned -> ds_load_b64.
    v2f a;
    a.x = arow[4 * kk];
    a.y = arow[4 * kk + 1];
    // B fragments: Z rows k0=4*kk+2*half and k0+1 at column c.
    //   m1 = kk>>2 (no carry across the 16-block), m2 = 4*(kk&3)+2*half(+1)
    const float u  = w1a[kk >> 2];
    const float v  = w1b[kk >> 2];
    v2f b0 = {u * w2a[2 * (kk & 3)], u * w2a[2 * (kk & 3) + 1]};
    v2f b1 = {v * w2b[2 * (kk & 3)], v * w2b[2 * (kk & 3) + 1]};
    acc0 = __builtin_amdgcn_wmma_f32_16x16x4_f32(false, a, false, b0,
                                                 (short)0, acc0, false, false);
    acc1 = __builtin_amdgcn_wmma_f32_16x16x4_f32(false, a, false, b1,
                                                 (short)0, acc1, false, false);
  }

  // D 16x16: vgpr v holds row M = v + 8*half at column c.
  float* po0 = out + ((size_t)n0 * NUM_ORDERS + 8 * half) * C_DIM + c;
  float* po1 = out + ((size_t)n1 * NUM_ORDERS + 8 * half) * C_DIM + c;
#pragma unroll
  for (int v = 0; v < 8; ++v) {
    po0[v * C_DIM] = acc0[v];
    po1[v * C_DIM] = acc1[v];
  }
}

// ---------------------------------------------------------------------------
// setup_inputs order: x1, x2, CG_vals, M1, M2, Mseg
// ---------------------------------------------------------------------------
extern "C" void kernel_launch(void* const* d_in, const int* in_sizes, int n_in,
                              void* d_out, int out_size, void* d_ws, size_t ws_size,
                              hipStream_t stream) {
  const float* x1   = (const float*)d_in[0];
  const float* x2   = (const float*)d_in[1];
  const float* vals = (const float*)d_in[2];
  const int* M1     = (const int*)d_in[3];
  const int* M2     = (const int*)d_in[4];
  const int* Mseg   = (const int*)d_in[5];
  float* A          = (float*)d_ws;  // 16*256*4 = 16 KB scratch
  float* out        = (float*)d_out;

  const int nnz     = in_sizes[2];
  const int n_batch = in_sizes[0] / (NUM_ORDERS * C_DIM);

  build_cg_dense<<<1, 512, 0, stream>>>(vals, M1, M2, Mseg, A, nnz);

  const int pairs  = (n_batch + 1) / 2;
  const int waves  = pairs * 4;        // one wave per (row-pair, channel tile)
  const int blocks = (waves + 7) / 8;  // 8 waves (256 threads) per block
  tp_wmma_kernel<<<blocks, 256, 0, stream>>>(x1, x2, A, out, n_batch);
}